// AdjEnsemble_21706764714067
// MI455X (gfx1250) — compile-verified
//
#include <hip/hip_runtime.h>

typedef __attribute__((ext_vector_type(2))) float v2f;
typedef __attribute__((ext_vector_type(8))) float v8f;

#define NN   8192
#define DD   64
#define KENS 4
#define NOUT 256
#define ROWS 16      // adj rows per workgroup
#define CK   128     // adj columns per phase-2 chunk
#define NCH  (NN / CK)
#define PSTR 132     // LDS row stride for P tile (16B-aligned rows, conflict-free)
#define PBUF (ROWS * PSTR)
#define FSTR 68      // LDS row stride for feature tile

__global__ __launch_bounds__(128) void AdjEnsemble_kernel(
    const float* __restrict__ sem,   // [N, D]   semantic_vec
    const float* __restrict__ adj,   // [N, N]
    const float* __restrict__ ctr,   // [K, D]   field_centers
    const float* __restrict__ fcw,   // [OUT, D]
    const float* __restrict__ fcb,   // [OUT]
    float*       __restrict__ out)   // [N, OUT]
{
    __shared__ float ldsP[2 * PBUF];           // double-buffered probability tile
    __shared__ float ldsF[ROWS * FSTR];
    __shared__ float ldsMax[ROWS];
    __shared__ float ldsSum[ROWS];

    const int t    = threadIdx.x;      // 0..127
    const int lane = t & 31;
    const int wave = t >> 5;           // 0..3
    const int hi   = lane >> 4;        // 0 or 1 (half-wave)
    const int l16  = lane & 15;
    const int m0   = blockIdx.x * ROWS;

    // ---------------- Phase 1: per-row max of x = -adj (no exp needed) ----------------
    {
        const int row = (t >> 3) & 15;   // 8 threads per row
        const int c8  = t & 7;
        const float* ap = adj + (size_t)(m0 + row) * NN;
        float mx = -3.402823466e+38f;
        for (int i = 0; i < NN / 32; ++i) {
            const float4 a = *(const float4*)(ap + 4 * (c8 + 8 * i));
            mx = fmaxf(mx, fmaxf(fmaxf(-a.x, -a.y), fmaxf(-a.z, -a.w)));
        }
        mx = fmaxf(mx, __shfl_xor(mx, 1));
        mx = fmaxf(mx, __shfl_xor(mx, 2));
        mx = fmaxf(mx, __shfl_xor(mx, 4));
        if ((lane & 7) == 0) ldsMax[row] = mx;
    }
    __syncthreads();

    // ---------------- Phase 2: S' = exp(-adj - m) @ sem (unnormalized), row sums -----
    v8f acc0 = {};                     // this wave owns D-columns [16*wave, 16*wave+16)
    v8f acc1 = {};
    float psum = 0.0f;
    const int   prow = (t >> 3) & 15;
    const int   pc8  = t & 7;
    const float prm  = ldsMax[prow];
    const int   n0   = wave * 16;

    // Build one chunk of the probability tile into pbuf (reads hit L2 after phase 1).
    auto buildP = [&](int k0, float* pbuf) {
        const float* ap = adj + (size_t)(m0 + prow) * NN + k0;
        float* pp = pbuf + prow * PSTR;
        #pragma unroll
        for (int j = 0; j < CK / 32; ++j) {
            const int col = 4 * (pc8 + 8 * j);
            const float4 a = *(const float4*)(ap + col);
            float4 p;
            p.x = __expf(-a.x - prm);
            p.y = __expf(-a.y - prm);
            p.z = __expf(-a.z - prm);
            p.w = __expf(-a.w - prm);
            psum += (p.x + p.y) + (p.z + p.w);
            *(float4*)(pp + col) = p;
        }
    };

    // WMMA over one chunk: D[16,16] += A[16,4] * B[4,16], K stepped by 8 (2 acc chains)
    auto consumeP = [&](const float* pbuf, int k0) {
        const float* pa0 = pbuf + l16 * PSTR + 2 * hi;
        const float* vb  = sem + (size_t)(k0 + 2 * hi) * DD + n0 + l16;
        #pragma unroll 4
        for (int kk = 0; kk < CK; kk += 8) {
            v2f a, b;
            const float* pa = pa0 + kk;
            const float* pb = vb + (size_t)kk * DD;
            a.x = pa[0];  a.y = pa[1];
            b.x = pb[0];  b.y = pb[DD];
            acc0 = __builtin_amdgcn_wmma_f32_16x16x4_f32(
                       false, a, false, b, (short)0, acc0, false, false);
            pa += 4; pb += 4 * DD;
            a.x = pa[0];  a.y = pa[1];
            b.x = pb[0];  b.y = pb[DD];
            acc1 = __builtin_amdgcn_wmma_f32_16x16x4_f32(
                       false, a, false, b, (short)0, acc1, false, false);
        }
    };

    // Prologue: build chunk 0, then pipeline: build c+1 while consuming c.
    buildP(0, ldsP);
    __syncthreads();
    for (int c = 0; c < NCH; ++c) {
        if (c + 1 < NCH) buildP((c + 1) * CK, ldsP + ((c + 1) & 1) * PBUF);
        consumeP(ldsP + (c & 1) * PBUF, c * CK);
        __syncthreads();
    }

    // Reduce per-row softmax denominators (8 threads per row -> lane 0 of group)
    psum += __shfl_xor(psum, 1);
    psum += __shfl_xor(psum, 2);
    psum += __shfl_xor(psum, 4);
    if ((lane & 7) == 0) ldsSum[prow] = psum;
    __syncthreads();

    // ---------------- Phase 3: features = 1/K * sum_k leaky(S - c_k) ----------------
    {
        const int d  = n0 + l16;
        const float c0 = ctr[0 * DD + d];
        const float c1 = ctr[1 * DD + d];
        const float c2 = ctr[2 * DD + d];
        const float c3 = ctr[3 * DD + d];
        #pragma unroll
        for (int r = 0; r < 8; ++r) {
            const int row = r + 8 * hi;
            const float s = (acc0[r] + acc1[r]) / ldsSum[row];
            float f = 0.0f, e;
            e = s - c0; f += (e >= 0.0f) ? e : 0.01f * e;
            e = s - c1; f += (e >= 0.0f) ? e : 0.01f * e;
            e = s - c2; f += (e >= 0.0f) ? e : 0.01f * e;
            e = s - c3; f += (e >= 0.0f) ? e : 0.01f * e;
            ldsF[row * FSTR + d] = 0.25f * f;
        }
    }
    __syncthreads();

    // ---------------- Phase 4: out = relu(F[16,64] @ W^T + b), WMMA again -----------
    {
        v8f oa0 = {}, oa1 = {}, oa2 = {}, oa3 = {};
        const float* fr = ldsF + l16 * FSTR + 2 * hi;
        const float* wb = fcw + (size_t)(64 * wave + l16) * DD + 2 * hi;
        #pragma unroll
        for (int ks = 0; ks < DD / 4; ++ks) {
            v2f a;
            const float* pa = fr + 4 * ks;
            a.x = pa[0];  a.y = pa[1];
            v2f b;
            const float* pb = wb + 4 * ks;
            b.x = pb[0]; b.y = pb[1];
            oa0 = __builtin_amdgcn_wmma_f32_16x16x4_f32(false, a, false, b, (short)0, oa0, false, false);
            pb += (size_t)16 * DD;
            b.x = pb[0]; b.y = pb[1];
            oa1 = __builtin_amdgcn_wmma_f32_16x16x4_f32(false, a, false, b, (short)0, oa1, false, false);
            pb += (size_t)16 * DD;
            b.x = pb[0]; b.y = pb[1];
            oa2 = __builtin_amdgcn_wmma_f32_16x16x4_f32(false, a, false, b, (short)0, oa2, false, false);
            pb += (size_t)16 * DD;
            b.x = pb[0]; b.y = pb[1];
            oa3 = __builtin_amdgcn_wmma_f32_16x16x4_f32(false, a, false, b, (short)0, oa3, false, false);
        }
        // bias + relu + store
        v8f oacc[4] = {oa0, oa1, oa2, oa3};
        #pragma unroll
        for (int tt = 0; tt < 4; ++tt) {
            const int o   = 64 * wave + 16 * tt + l16;
            const float bias = fcb[o];
            #pragma unroll
            for (int r = 0; r < 8; ++r) {
                const int row = r + 8 * hi;
                out[(size_t)(m0 + row) * NOUT + o] = fmaxf(oacc[tt][r] + bias, 0.0f);
            }
        }
    }
}

extern "C" void kernel_launch(void* const* d_in, const int* in_sizes, int n_in,
                              void* d_out, int out_size, void* d_ws, size_t ws_size,
                              hipStream_t stream) {
    (void)in_sizes; (void)n_in; (void)out_size; (void)d_ws; (void)ws_size;
    const float* sem = (const float*)d_in[0];   // [8192, 64]
    const float* adj = (const float*)d_in[1];   // [8192, 8192]
    const float* ctr = (const float*)d_in[2];   // [4, 64]
    const float* fcw = (const float*)d_in[3];   // [256, 64]
    const float* fcb = (const float*)d_in[4];   // [256]
    float* out = (float*)d_out;                 // [8192, 256]

    dim3 grid(NN / ROWS);   // 512 workgroups, 16 rows each
    dim3 block(128);        // 4 wave32s
    AdjEnsemble_kernel<<<grid, block, 0, stream>>>(sem, adj, ctr, fcw, fcb, out);
}